// CapsuleFC_1709396984015
// MI455X (gfx1250) — compile-verified
//
#include <hip/hip_runtime.h>

// Problem constants (from the reference)
#define B_SZ   128
#define N_SZ   4096
#define M_SZ   64
#define D_SZ   16
#define NCHUNK 128   // n's per workgroup (processed 2 at a time)
#define BTILE  4     // b's per workgroup -> 16 GEMM rows = one WMMA row tile
#define LN_EPS 1e-5f

typedef float v2f __attribute__((ext_vector_type(2)));
typedef float v8f __attribute__((ext_vector_type(8)));

// ---------------------------------------------------------------------------
// Kernel 0: zero the accumulation workspace (harness poisons it with 0xAA)
// ---------------------------------------------------------------------------
__global__ __launch_bounds__(256) void caps_zero(float* __restrict__ ws, int n) {
    int i = blockIdx.x * blockDim.x + threadIdx.x;
    if (i < n) ws[i] = 0.0f;
}

// ---------------------------------------------------------------------------
// Kernel 1: fused vote-GEMM (WMMA f32 16x16x4) + routing softmax + aggregation
// Grid: (N/NCHUNK, B/BTILE), Block: 256 threads = 8 wave32
// Per 2-n iteration: 4 WMMAs/wave, 4 workgroup barriers, wave-level softmax.
// ---------------------------------------------------------------------------
__global__ __launch_bounds__(256) void caps_main(const float* __restrict__ input,
                                                 const float* __restrict__ ncv,
                                                 const float* __restrict__ w,
                                                 float* __restrict__ ws) {
    __shared__ float u_lds[2 * 16 * 256];   // votes u[q][(b,a),(d,m)]  32 KB
    __shared__ float red_max[8 * 2];        // per-(wave,q) partial maxima
    __shared__ float red_sum[8 * 2];        // per-(wave,q) partial sums

    const int tid = threadIdx.x;
    const int b0  = blockIdx.y * BTILE;
    const int n0  = blockIdx.x * NCHUNK;

    // WMMA lane decomposition (wave32)
    const int lane = tid & 31;
    const int wave = tid >> 5;        // 0..7; wave handles column tiles 2w, 2w+1
    const int half = lane >> 4;       // 0: K=0,1 / rows 0-7 ; 1: K=2,3 / rows 8-15
    const int r16  = lane & 15;
    const int bl   = r16 >> 2;        // local b of the A row (row = bl*4 + ai)
    const int ai   = r16 & 3;         // 'a' index of the pose row
    const int x0   = half * 2;        // first K ('x') element this half holds

    // qk / aggregation decomposition: thread <-> (b, m).
    // All 32 lanes of a wave share one b; m = (wave&1)*32 + lane.
    const int bq  = tid >> 6;         // 0..3
    const int mq  = tid & 63;         // 0..63
    const int wpar = wave & 1;        // which half of the m-row this wave owns

    // next_capsule_value row for this thread's (b, m): loop-invariant -> registers
    float ncv_reg[D_SZ];
    {
        const float* src = ncv + ((size_t)(b0 + bq) * M_SZ + mq) * D_SZ;
#pragma unroll
        for (int k = 0; k < D_SZ; ++k) ncv_reg[k] = src[k];
    }

    float acc[D_SZ];
#pragma unroll
    for (int k = 0; k < D_SZ; ++k) acc[k] = 0.0f;

    for (int ni = 0; ni < NCHUNK; ni += 2) {
        // ================= vote GEMM: u = A @ W (two n's) =================
#pragma unroll
        for (int q = 0; q < 2; ++q) {
            const int n = n0 + ni + q;

            // A operand: 16x4 tile of input poses, row=(b,a), col=x
            const float* ap = input + ((size_t)(b0 + bl) * N_SZ + n) * D_SZ + ai * 4 + x0;
            v2f av; av.x = ap[0]; av.y = ap[1];

            const float* wn = w + (size_t)n * (4 * 4 * M_SZ);
            // Prefetch w two n's ahead while we compute
            if (ni + 2 + q < NCHUNK)
                __builtin_prefetch(wn + 2 * (4 * 4 * M_SZ) + tid * 4, 0, 3);

#pragma unroll
            for (int jj = 0; jj < 2; ++jj) {
                const int j   = wave * 2 + jj;      // column tile 0..15
                const int col = j * 16 + r16;       // global column = d*64 + m
                const int dd  = col >> 6;
                const int mm  = col & 63;
                const float* bp = wn + x0 * (4 * M_SZ) + dd * M_SZ + mm;
                v2f bv; bv.x = bp[0]; bv.y = bp[4 * M_SZ];   // rows x0, x0+1

                v8f c = {};
                v8f u = __builtin_amdgcn_wmma_f32_16x16x4_f32(
                    false, av, false, bv, (short)0, c, false, false);

#pragma unroll
                for (int r = 0; r < 8; ++r)
                    u_lds[q * 4096 + (r + 8 * half) * 256 + col] = u[r];
            }
        }
        __syncthreads();

        // ============ qk scores (u cached in registers for reuse) ============
        float uq[2][D_SZ];
        float qk[2];
#pragma unroll
        for (int q = 0; q < 2; ++q) {
            float s = 0.0f;
#pragma unroll
            for (int a = 0; a < 4; ++a)
#pragma unroll
                for (int d = 0; d < 4; ++d) {
                    const float uv = u_lds[q * 4096 + (bq * 4 + a) * 256 + d * 64 + mq];
                    uq[q][a * 4 + d] = uv;
                    s += uv * ncv_reg[a * 4 + d];
                }
            qk[q] = s * 0.25f;   // 1/sqrt(D), D=16
        }

        // ============ softmax over m: wave butterfly + 2-slot combine ========
        float mx0 = qk[0], mx1 = qk[1];
#pragma unroll
        for (int off = 16; off > 0; off >>= 1) {
            mx0 = fmaxf(mx0, __shfl_xor(mx0, off, 32));
            mx1 = fmaxf(mx1, __shfl_xor(mx1, off, 32));
        }
        if (lane == 0) {
            red_max[(bq * 2 + wpar) * 2 + 0] = mx0;
            red_max[(bq * 2 + wpar) * 2 + 1] = mx1;
        }
        __syncthreads();
        const float rmax0 = fmaxf(red_max[(bq * 2) * 2 + 0], red_max[(bq * 2 + 1) * 2 + 0]);
        const float rmax1 = fmaxf(red_max[(bq * 2) * 2 + 1], red_max[(bq * 2 + 1) * 2 + 1]);

        const float e0 = __expf(qk[0] - rmax0);
        const float e1 = __expf(qk[1] - rmax1);
        float s0 = e0, s1 = e1;
#pragma unroll
        for (int off = 16; off > 0; off >>= 1) {
            s0 += __shfl_xor(s0, off, 32);
            s1 += __shfl_xor(s1, off, 32);
        }
        if (lane == 0) {
            red_sum[(bq * 2 + wpar) * 2 + 0] = s0;
            red_sum[(bq * 2 + wpar) * 2 + 1] = s1;
        }
        __syncthreads();
        const float rsum0 = red_sum[(bq * 2) * 2 + 0] + red_sum[(bq * 2 + 1) * 2 + 0];
        const float rsum1 = red_sum[(bq * 2) * 2 + 1] + red_sum[(bq * 2 + 1) * 2 + 1];

        // softmax, then the reference's divide-by-(sum(softmax)+1e-10)
        const float p0 = e0 / (rsum0 * (1.0f + 1e-10f) + 1e-10f);
        const float p1 = e1 / (rsum1 * (1.0f + 1e-10f) + 1e-10f);

        // ============ out[b,m,:] += p * u (register accumulators) ============
#pragma unroll
        for (int k = 0; k < D_SZ; ++k)
            acc[k] += p0 * uq[0][k] + p1 * uq[1][k];

        __syncthreads();   // u_lds / red_* reused next iteration
    }

    // Accumulate this n-chunk's contribution into the global workspace
    float* dst = ws + ((size_t)(b0 + bq) * M_SZ + mq) * D_SZ;
#pragma unroll
    for (int k = 0; k < D_SZ; ++k) atomicAdd(&dst[k], acc[k]);
}

// ---------------------------------------------------------------------------
// Kernel 2: LayerNorm over D=16 per (b, m) row
// ---------------------------------------------------------------------------
__global__ __launch_bounds__(256) void caps_ln(const float* __restrict__ ws,
                                               const float* __restrict__ ln_w,
                                               const float* __restrict__ ln_b,
                                               float* __restrict__ out) {
    int row = blockIdx.x * blockDim.x + threadIdx.x;
    if (row >= B_SZ * M_SZ) return;
    const float* src = ws + (size_t)row * D_SZ;
    float v[D_SZ];
    float s = 0.0f;
#pragma unroll
    for (int k = 0; k < D_SZ; ++k) { v[k] = src[k]; s += v[k]; }
    const float mu = s * (1.0f / D_SZ);
    float var = 0.0f;
#pragma unroll
    for (int k = 0; k < D_SZ; ++k) { float d = v[k] - mu; var += d * d; }
    var *= (1.0f / D_SZ);
    const float inv = rsqrtf(var + LN_EPS);
    float* dst = out + (size_t)row * D_SZ;
#pragma unroll
    for (int k = 0; k < D_SZ; ++k)
        dst[k] = (v[k] - mu) * inv * ln_w[k] + ln_b[k];
}

// ---------------------------------------------------------------------------
extern "C" void kernel_launch(void* const* d_in, const int* in_sizes, int n_in,
                              void* d_out, int out_size, void* d_ws, size_t ws_size,
                              hipStream_t stream) {
    (void)in_sizes; (void)n_in; (void)out_size; (void)ws_size;
    const float* input = (const float*)d_in[0];   // [B, N, D]
    const float* ncv   = (const float*)d_in[1];   // [B, M, D]
    const float* w     = (const float*)d_in[2];   // [N, 4, 4, M]
    const float* ln_w  = (const float*)d_in[3];   // [D]
    const float* ln_b  = (const float*)d_in[4];   // [D]
    // d_in[5] = num_iter (== 1): the reference performs a single pass.
    float* out = (float*)d_out;
    float* ws  = (float*)d_ws;                    // B*M*D floats = 512 KB accumulator

    const int nacc = B_SZ * M_SZ * D_SZ;
    caps_zero<<<(nacc + 255) / 256, 256, 0, stream>>>(ws, nacc);

    dim3 grid(N_SZ / NCHUNK, B_SZ / BTILE);       // (32, 32)
    caps_main<<<grid, 256, 0, stream>>>(input, ncv, w, ws);

    caps_ln<<<(B_SZ * M_SZ + 255) / 256, 256, 0, stream>>>(ws, ln_w, ln_b, out);
}